// UnrolledFBS_72421738545771
// MI455X (gfx1250) — compile-verified
//
#include <hip/hip_runtime.h>
#include <math.h>

// ---------------------------------------------------------------------------
// UnrolledFBS on MI455X (gfx1250): single persistent kernel, fp32 WMMA GEMVs.
// ---------------------------------------------------------------------------

#define NWG   32
#define NT    256
#define SZP   256          // SIZE*SIZE
#define NPIX  2304         // N_CH * SIZE * SIZE
#define INF_  9218
#define HID   256
#define OUTF  4612
#define TITER 20
#define EPSC      1e-8f
#define SAFE_EPSC 1e-12f
#define TAUC      0.1f
#define GMAXC     0.8463922f

typedef __attribute__((ext_vector_type(2))) float v2f;
typedef __attribute__((ext_vector_type(8))) float v8f;

// workspace layout (float offsets)
#define BUFSZ    9248
#define OFF_BUF0 0
#define OFF_BUF1 (OFF_BUF0 + BUFSZ)
#define OFF_OUT  (OFF_BUF1 + BUFSZ)       // 4612 (pad 4640)
#define OFF_SA   (OFF_OUT + 4640)         // 32 WGs * 4 sums
#define OFF_SB   (OFF_SA + 128)           // 16 k-slices * 256 h1 partials
#define OFF_SC   (OFF_SB + 4096)          // 32 WGs * 2 (u2,v2)
#define OFF_BAR  (OFF_SC + 64)            // 8 ints for grid barrier

__device__ __forceinline__ float softplusf(float x) {
    return fmaxf(x, 0.f) + log1pf(expf(-fabsf(x)));
}
__device__ __forceinline__ float sigmoidf(float x) {
    return 1.f / (1.f + expf(-x));
}

__device__ __forceinline__ void grid_sync(int* bar) {
    __syncthreads();
    if (threadIdx.x == 0) {
        __threadfence();
        int* cnt = bar;
        int* gen = bar + 1;
        int g = __hip_atomic_load(gen, __ATOMIC_RELAXED, __HIP_MEMORY_SCOPE_AGENT);
        int prev = __hip_atomic_fetch_add(cnt, 1, __ATOMIC_ACQ_REL, __HIP_MEMORY_SCOPE_AGENT);
        if (prev == NWG - 1) {
            __hip_atomic_store(cnt, 0, __ATOMIC_RELAXED, __HIP_MEMORY_SCOPE_AGENT);
            __threadfence();
            __hip_atomic_fetch_add(gen, 1, __ATOMIC_ACQ_REL, __HIP_MEMORY_SCOPE_AGENT);
        } else {
            while (__hip_atomic_load(gen, __ATOMIC_ACQUIRE, __HIP_MEMORY_SCOPE_AGENT) == g) {
                __builtin_amdgcn_s_sleep(1);
            }
        }
        __threadfence();
    }
    __syncthreads();
}

__device__ __forceinline__ float block_reduce(float v, float* red) {
    int tid = threadIdx.x;
    red[tid] = v;
    __syncthreads();
    for (int s = NT / 2; s > 0; s >>= 1) {
        if (tid < s) red[tid] += red[tid + s];
        __syncthreads();
    }
    float r = red[0];
    __syncthreads();
    return r;
}

__global__ void fbs_init_kernel(int* bar) {
    if (threadIdx.x < 8) bar[threadIdx.x] = 0;
}

__global__ __launch_bounds__(NT) void fbs_persistent_kernel(
    const float* __restrict__ noisy,
    const float* __restrict__ W1, const float* __restrict__ b1,
    const float* __restrict__ W2, const float* __restrict__ b2,
    const float* __restrict__ W3, const float* __restrict__ b3,
    float* __restrict__ out_final, float* __restrict__ ws)
{
    __shared__ float h1L[HID];
    __shared__ float h2L[HID];
    __shared__ float red[NT];

    const int tid   = threadIdx.x;
    const int wg    = blockIdx.x;
    const int gt    = wg * NT + tid;
    const int lane  = tid & 31;
    const int wv    = tid >> 5;
    const int gwave = wg * 8 + wv;
    const int half  = lane >> 4;       // 0: lanes 0-15, 1: lanes 16-31
    const int c0    = half * 2;        // K sub-offset per ISA A/B layout
    const int ln    = lane & 15;

    float* buf0 = ws + OFF_BUF0;
    float* buf1 = ws + OFF_BUF1;
    float* outb = ws + OFF_OUT;
    float* sA   = ws + OFF_SA;
    float* sB   = ws + OFF_SB;
    float* sC   = ws + OFF_SC;
    int*   bar  = (int*)(ws + OFF_BAR);

    // ---- init: buf1 holds x = y_prev = p_prev = z_prev = noisy_pad ----
    for (int j = gt; j < 4 * NPIX; j += NWG * NT) {
        int i = j % NPIX;
        buf1[j] = (i < SZP) ? noisy[i] : 0.f;
    }

    // loop-carried uniform scalars
    float lam = 1.0f, mu = 0.0f, gamma = 0.5f, gamma_prev = 0.5f;
    float scale = 0.0f, su2 = 0.0f, sv2 = 0.0f, delta_prev = 0.0f;

    grid_sync(bar);

#pragma unroll 1
    for (int n = 0; n < TITER; ++n) {
        // ---- scalar/safeguard update from previous iteration (redundant per thread)
        if (n > 0) {
            float s0 = outb[4608], s1 = outb[4609], s2 = outb[4610];
            lam = softplusf(s0);
            gamma = GMAXC * sigmoidf(s1);
            mu = softplusf(s2);
            gamma_prev = gamma;     // reference sets gamma_prev = new gamma
            float sumU = 0.f, sumV = 0.f;
            for (int w = 0; w < NWG; ++w) { sumU += sC[2 * w]; sumV += sC[2 * w + 1]; }
            float radius = sqrtf(fmaxf(delta_prev, 0.f) + SAFE_EPSC);
            float nrm = sqrtf(sumU + sumV);
            scale = fminf(radius / (nrm + SAFE_EPSC), 1.f);
            su2 = scale * scale * sumU;
            sv2 = scale * scale * sumV;
        }
        float a_n  = 1.f / (1.f + mu);
        float ab_n = a_n * gamma / (gamma_prev + EPSC);
        float th   = gamma * lam;
        float th_h = th + mu;
        float th_b = th;                          // BETA_BAR = 1
        float cz   = th_b * gamma / (th_h + EPSC);

        float* prevB = (n & 1) ? buf0 : buf1;
        float* curB  = (n & 1) ? buf1 : buf0;

        // ---- Phase A: elementwise one_step (2304 elems, one per thread) ----
        float p_res = 0.f, p1 = 0.f, p2 = 0.f, p3 = 0.f;
        if (gt < NPIX) {
            int i = gt;
            float np_i = (i < SZP) ? noisy[i] : 0.f;
            float u_i = (n == 0) ? 0.f : scale * outb[i];
            float v_i = (n == 0) ? 0.f : scale * outb[NPIX + i];
            float xv  = prevB[i];
            float ppv = prevB[NPIX + i];
            float ypv = prevB[2 * NPIX + i];
            float zpv = prevB[3 * NPIX + i];
            float y  = xv + a_n * (ypv - xv) + u_i;
            float z  = xv + a_n * (ppv - xv) + ab_n * (zpv - ppv) + cz * u_i + v_i;
            float Cy = y - np_i;
            float t  = z - gamma * Cy;
            float at = fabsf(t) - gamma * TAUC;
            float pr = (at > 0.f) ? copysignf(at, t) : 0.f;
            float xn = xv + lam * (pr - z) + ab_n * lam * (zpv - ppv);
            curB[i]            = xn;
            curB[NPIX + i]     = pr;
            curB[2 * NPIX + i] = y;
            curB[3 * NPIX + i] = z;
            float d;
            d = pr - y;   p_res = d * d;
            d = pr - xn;  p1 = d * d;
            d = y - ypv;  p2 = d * d;
            d = z - zpv;  p3 = d * d;
        }
        float r;
        r = block_reduce(p_res, red); if (tid == 0) sA[wg * 4 + 0] = r;
        r = block_reduce(p1, red);    if (tid == 0) sA[wg * 4 + 1] = r;
        r = block_reduce(p2, red);    if (tid == 0) sA[wg * 4 + 2] = r;
        r = block_reduce(p3, red);    if (tid == 0) sA[wg * 4 + 3] = r;

        grid_sync(bar);   // sync 1

        // ---- redundant global reduce -> residual + delta ----
        float sres = 0.f, sd1 = 0.f, sd2 = 0.f, sd3 = 0.f;
        for (int w = 0; w < NWG; ++w) {
            sres += sA[w * 4 + 0]; sd1 += sA[w * 4 + 1];
            sd2  += sA[w * 4 + 2]; sd3 += sA[w * 4 + 3];
        }
        float res_n = sqrtf(sres + EPSC);
        float delta = gamma * sd1 + lam * sd2 + mu * sd3 + th_h * su2 + th_b * sv2;
        if (!isfinite(delta)) delta = 0.f;
        if (wg == 0 && tid == 0) out_final[NPIX + n] = res_n;
        delta_prev = delta;

        if (n == TITER - 1) break;   // last iteration only needs pr + residual

        float tctx = (float)n / (float)TITER;

        // ---- GEMV1 (9218x256) via V_WMMA_F32_16X16X4_F32, K split 16 ways ----
        {
            int ct  = gwave & 15;           // column tile (16 cols)
            int ks  = gwave >> 4;           // k slice (576 rows each)
            int col = ct * 16 + ln;
            int kbase = ks * 576;
            // loop-invariant per-lane bases -> immediate-offset addressing
            const float* ap = curB + kbase + c0;
            const float* bp = W1 + (kbase + c0) * HID + col;
            v8f acc = {};
            for (int kb = 0; kb < 144; ++kb) {
                v2f a = *(const v2f*)(ap + kb * 4);   // vector replicated over rows
                v2f b;
                b.x = bp[kb * 4 * HID];
                b.y = bp[kb * 4 * HID + HID];
                acc = __builtin_amdgcn_wmma_f32_16x16x4_f32(
                    false, a, false, b, (short)0, acc, false, false);
            }
            if (half == 0) sB[ks * HID + col] = acc[0];       // row 0 of D
        }
        grid_sync(bar);   // sync 2

        // ---- h1 = relu(b1 + sum of 16 partials + context tail) (redundant/WG) ----
        {
            int j = tid;
            float s = b1[j];
            for (int ksl = 0; ksl < 16; ++ksl) s += sB[ksl * HID + j];
            s += delta * W1[9216 * HID + j] + tctx * W1[9217 * HID + j];
            h1L[j] = fmaxf(s, 0.f);
        }
        __syncthreads();

        // ---- GEMV2 (256x256) WMMA, A from LDS, 16 tiles over 8 waves ----
        for (int tt = 0; tt < 2; ++tt) {
            int ct  = wv + tt * 8;
            int col = ct * 16 + ln;
            const float* bp = W2 + c0 * HID + col;
            v8f acc = {};
            for (int kb = 0; kb < 64; ++kb) {
                int k = kb * 4;
                v2f a;
                a.x = h1L[k + c0];
                a.y = h1L[k + c0 + 1];
                v2f b;
                b.x = bp[kb * 4 * HID];
                b.y = bp[kb * 4 * HID + HID];
                acc = __builtin_amdgcn_wmma_f32_16x16x4_f32(
                    false, a, false, b, (short)0, acc, false, false);
            }
            if (half == 0) h2L[col] = fmaxf(acc[0] + b2[col], 0.f);
        }
        __syncthreads();

        // ---- GEMV3 (256x4612) WMMA: 288 column tiles over 256 waves + 4 tail ----
        float pu2 = 0.f, pv2 = 0.f;
        for (int rep = 0; rep < 2; ++rep) {
            int ct = gwave + rep * 256;
            if (ct >= 288) break;            // wave-uniform
            int col = ct * 16 + ln;
            const float* bp = W3 + c0 * OUTF + col;
            v8f acc = {};
            for (int kb = 0; kb < 64; ++kb) {
                int k = kb * 4;
                v2f a;
                a.x = h2L[k + c0];
                a.y = h2L[k + c0 + 1];
                v2f b;
                b.x = bp[kb * 4 * OUTF];
                b.y = bp[kb * 4 * OUTF + OUTF];
                acc = __builtin_amdgcn_wmma_f32_16x16x4_f32(
                    false, a, false, b, (short)0, acc, false, false);
            }
            if (half == 0) {
                float d = acc[0] + b3[col];
                if (!isfinite(d)) d = 0.f;   // nan_to_num on u_raw/v_raw maps
                outb[col] = d;
                if (ct < 144) pu2 += d * d; else pv2 += d * d;
            }
        }
        // tail columns 4608..4611 (the s scalars), scalar loop by WG0
        if (wg == 0 && tid < 4) {
            int col = 4608 + tid;
            float s = b3[col];
            for (int k = 0; k < HID; ++k) s += h2L[k] * W3[k * OUTF + col];
            outb[col] = s;
        }
        float ru = block_reduce(pu2, red);
        float rv = block_reduce(pv2, red);
        if (tid == 0) { sC[wg * 2] = ru; sC[wg * 2 + 1] = rv; }

        grid_sync(bar);   // sync 3
    }

    // ---- final output: p_prev = pr of iteration 19 (curB == buf1) ----
    for (int i = gt; i < NPIX; i += NWG * NT) {
        out_final[i] = buf1[NPIX + i];
    }
}

extern "C" void kernel_launch(void* const* d_in, const int* in_sizes, int n_in,
                              void* d_out, int out_size, void* d_ws, size_t ws_size,
                              hipStream_t stream) {
    const float* noisy = (const float*)d_in[0];
    const float* W1    = (const float*)d_in[1];
    const float* b1    = (const float*)d_in[2];
    const float* W2    = (const float*)d_in[3];
    const float* b2    = (const float*)d_in[4];
    const float* W3    = (const float*)d_in[5];
    const float* b3    = (const float*)d_in[6];
    float* out = (float*)d_out;
    float* ws  = (float*)d_ws;
    int* bar = (int*)(ws + OFF_BAR);

    fbs_init_kernel<<<1, 32, 0, stream>>>(bar);
    fbs_persistent_kernel<<<NWG, NT, 0, stream>>>(noisy, W1, b1, W2, b2, W3, b3, out, ws);
}